// IDR_DDC_16655883174450
// MI455X (gfx1250) — compile-verified
//
#include <hip/hip_runtime.h>

typedef __attribute__((ext_vector_type(2))) float v2f;
typedef __attribute__((ext_vector_type(8))) float v8f;

#define B_  16
#define C_  64
#define H_  192
#define W_  192
#define KK_ 576   // C * 3 * 3
#define PLANE (H_ * W_)

// ---------------------------------------------------------------------------
// Kernel A: hid[b][c] = leaky_relu( sum_k d[b][k] * w1[c][k], slope=0.1 )
// 1 block x 1024 threads, one element per thread (K=64, float4 loads).
// ---------------------------------------------------------------------------
__global__ __launch_bounds__(1024) void hid_kernel(const float* __restrict__ d,
                                                   const float* __restrict__ w1,
                                                   float* __restrict__ hid) {
  int t = threadIdx.x;            // 0..1023
  int b = t >> 6;
  int c = t & 63;
  const float4* dr = (const float4*)(d  + b * 64);
  const float4* wr = (const float4*)(w1 + c * 64);
  float acc = 0.f;
#pragma unroll
  for (int i = 0; i < 16; ++i) {
    float4 a = dr[i];
    float4 w = wr[i];
    acc += a.x * w.x + a.y * w.y + a.z * w.z + a.w * w.w;
  }
  hid[t] = (acc > 0.f) ? acc : 0.1f * acc;
}

// ---------------------------------------------------------------------------
// Kernel B: ker(16 x 576) = hid(16 x 64) @ w2^T  via V_WMMA_F32_16X16X4_F32.
// One wave per 16-wide N tile; 16 K-steps of K=4.
//   A frag (16x4 f32): lane m(0..15): VGPR0=A[m][k0], VGPR1=A[m][k0+1];
//                      lanes 16..31:  K offsets +2/+3.
//   B frag (4x16 f32): same pattern transposed; B[k][n] = w2[n*64 + k].
//   D (16x16 f32, 8 VGPRs): VGPR r -> M=r (lanes 0-15) / M=r+8 (lanes 16-31).
// ---------------------------------------------------------------------------
__global__ __launch_bounds__(32) void kergen_kernel(const float* __restrict__ hid,
                                                    const float* __restrict__ w2,
                                                    float* __restrict__ ker) {
  int lane = threadIdx.x;               // 0..31
  int m    = lane & 15;
  int koff = (lane < 16) ? 0 : 2;
  int ncol = blockIdx.x * 16 + m;       // 0..575

  const float* arow = hid + m * 64;     // A[m][*]
  const float* brow = w2 + (size_t)ncol * 64;  // B[k][ncol] = w2[ncol][k]

  v8f acc = {};
#pragma unroll
  for (int s = 0; s < 16; ++s) {
    int k0 = 4 * s + koff;
    v2f a; a.x = arow[k0]; a.y = arow[k0 + 1];
    v2f b; b.x = brow[k0]; b.y = brow[k0 + 1];
    acc = __builtin_amdgcn_wmma_f32_16x16x4_f32(
        /*neg_a=*/false, a, /*neg_b=*/false, b,
        /*c_mod=*/(short)0, acc, /*reuse_a=*/false, /*reuse_b=*/false);
  }

  int mbase = (lane < 16) ? 0 : 8;
#pragma unroll
  for (int r = 0; r < 8; ++r) {
    ker[(size_t)(mbase + r) * KK_ + ncol] = acc[r];
  }
}

// ---------------------------------------------------------------------------
// Kernel C: per-(b,c) dynamic depthwise 3x3, zero pad=1.
// Each thread computes 4 horizontal pixels (float4 store).
// Grid: (36 blocks-per-plane, 1024 planes), block 256 threads.
// Per stencil row: one B128 load + two B32 edge loads.
// ---------------------------------------------------------------------------
__global__ __launch_bounds__(256) void conv_kernel(const float* __restrict__ x0,
                                                   const float* __restrict__ ker,
                                                   float* __restrict__ out) {
  int p = blockIdx.y;                         // plane = b*64 + c
  const float* xp = x0  + (size_t)p * PLANE;
  float*       op = out + (size_t)p * PLANE;
  const float* kp = ker + (size_t)(p >> 6) * KK_ + (size_t)(p & 63) * 9;

  float k00 = kp[0], k01 = kp[1], k02 = kp[2];
  float k10 = kp[3], k11 = kp[4], k12 = kp[5];
  float k20 = kp[6], k21 = kp[7], k22 = kp[8];

  int qid = blockIdx.x * 256 + threadIdx.x;   // 0..9215
  int y   = qid / 48;
  int xq  = (qid % 48) * 4;                   // 0,4,...,188

  float r[3][6];
#pragma unroll
  for (int i = 0; i < 3; ++i) {
    int yy = y + i - 1;
    if (yy < 0 || yy >= H_) {
#pragma unroll
      for (int j = 0; j < 6; ++j) r[i][j] = 0.f;
    } else {
      const float* row = xp + (size_t)yy * W_ + xq;
      float4 v = *(const float4*)row;
      r[i][0] = (xq == 0)       ? 0.f : row[-1];
      r[i][1] = v.x; r[i][2] = v.y; r[i][3] = v.z; r[i][4] = v.w;
      r[i][5] = (xq == W_ - 4)  ? 0.f : row[4];
    }
  }

  float o[4];
#pragma unroll
  for (int j = 0; j < 4; ++j) {
    o[j] = k00 * r[0][j] + k01 * r[0][j + 1] + k02 * r[0][j + 2]
         + k10 * r[1][j] + k11 * r[1][j + 1] + k12 * r[1][j + 2]
         + k20 * r[2][j] + k21 * r[2][j + 1] + k22 * r[2][j + 2];
  }
  float4 ov; ov.x = o[0]; ov.y = o[1]; ov.z = o[2]; ov.w = o[3];
  *(float4*)(op + (size_t)y * W_ + xq) = ov;
}

// ---------------------------------------------------------------------------
extern "C" void kernel_launch(void* const* d_in, const int* in_sizes, int n_in,
                              void* d_out, int out_size, void* d_ws, size_t ws_size,
                              hipStream_t stream) {
  const float* x0 = (const float*)d_in[0];   // (16,64,192,192)
  const float* d  = (const float*)d_in[1];   // (16,64)
  const float* w1 = (const float*)d_in[2];   // (64,64)
  const float* w2 = (const float*)d_in[3];   // (576,64)
  float* out = (float*)d_out;                // (16,64,192,192)

  float* hid = (float*)d_ws;                              // 16*64  floats (4 KB)
  float* ker = (float*)((char*)d_ws + 1024 * sizeof(float)); // 16*576 floats (36 KB)

  hid_kernel<<<1, 1024, 0, stream>>>(d, w1, hid);
  kergen_kernel<<<36, 32, 0, stream>>>(hid, w2, ker);
  conv_kernel<<<dim3(36, 1024), 256, 0, stream>>>(x0, ker, out);
}